// ZemlianovaRNN_64733747085450
// MI455X (gfx1250) — compile-verified
//
#include <hip/hip_runtime.h>

typedef float v2f __attribute__((ext_vector_type(2)));
typedef float v4f __attribute__((ext_vector_type(4)));
typedef float v8f __attribute__((ext_vector_type(8)));
typedef unsigned int v4u __attribute__((ext_vector_type(4)));
typedef int v4i __attribute__((ext_vector_type(4)));
typedef int v8i __attribute__((ext_vector_type(8)));

constexpr int HDIM   = 500;   // hidden
constexpr int HP     = 512;   // padded hidden
constexpr int BATCH  = 128;
constexpr int TSTEPS = 500;
constexpr int LDSROW = 136;   // 128 + 8 floats TDM pad -> bank-conflict-free
constexpr int NGEMM  = 32;    // 32 col-tile workgroups
constexpr int NBLK   = 33;    // + 1 output workgroup

// ---- grid-wide barrier: monotonic counter + agent-scope fences ----
__device__ __forceinline__ void grid_barrier(unsigned* bar, unsigned phase) {
  __builtin_amdgcn_fence(__ATOMIC_RELEASE, "agent");   // flush h stores to L2
  __syncthreads();
  if (threadIdx.x == 0) {
    __hip_atomic_fetch_add(bar, 1u, __ATOMIC_RELAXED, __HIP_MEMORY_SCOPE_AGENT);
    const unsigned tgt = phase * (unsigned)gridDim.x;
    while (__hip_atomic_load(bar, __ATOMIC_RELAXED, __HIP_MEMORY_SCOPE_AGENT) < tgt)
      __builtin_amdgcn_s_sleep(1);
  }
  __syncthreads();
  __builtin_amdgcn_fence(__ATOMIC_ACQUIRE, "agent");   // invalidate WGP$ (all waves)
}

// ---- TDM: DMA hT[512][128] f32 slab global -> LDS with 8-dword row padding ----
__device__ __forceinline__ void tdm_load_hT(unsigned lds_off, const float* src) {
  const unsigned long long ga = (unsigned long long)(uintptr_t)src;
  v4u g0;
  g0[0] = 1u;                                    // count=1, normal mode
  g0[1] = lds_off;                               // LDS byte address
  g0[2] = (unsigned)ga;                          // global_addr[31:0]
  g0[3] = (unsigned)((ga >> 32) & 0x1FFFFFFull)  // global_addr[56:32]
        | (2u << 30);                            // type=2 (image)
  v8i g1;
  g1[0] = (2 << 16)        // data_size = 4B
        | (1 << 20)        // pad_enable
        | (6 << 22)        // pad_interval: 128 dwords (one 512B row)
        | (7 << 25);       // pad_amount:   8 dwords (32B) -> 136-float stride
  g1[1] = (BATCH << 16);   // tensor_dim0 = 128
  g1[2] = (HP << 16);      // tensor_dim1 = 512
  g1[3] = (BATCH << 16);   // tile_dim0 = 128
  g1[4] = HP;              // tile_dim1 = 512, tile_dim2 = 0
  g1[5] = BATCH;           // tensor_dim0_stride = 128
  g1[6] = 0;
  g1[7] = 0;
  const v4i z4 = {0, 0, 0, 0};
  const v8i z8 = {0, 0, 0, 0, 0, 0, 0, 0};
  __builtin_amdgcn_tensor_load_to_lds(g0, g1, z4, z4, z8, 0);
}

__global__ __launch_bounds__(256, 1)
void zrnn_kernel(const float* __restrict__ input, const float* __restrict__ hidden,
                 const float* __restrict__ noise, const float* __restrict__ W_rec,
                 const float* __restrict__ W_in, const float* __restrict__ b_rec,
                 const float* __restrict__ W_out, const float* __restrict__ b_out,
                 float* __restrict__ out, unsigned* bar,
                 float* __restrict__ hT0, float* __restrict__ hT1) {
  __shared__ float lds_h[HP * LDSROW];   // 272 KB: transposed h slab (TDM dest)
  __shared__ v2f   lds_A[128 * 32];      //  32 KB: W_eff^T slice, WMMA-A lane layout
  __shared__ float lds_win0[16], lds_win1[16], lds_brec[16];
  __shared__ float lds_red[256];

  const int tid  = threadIdx.x;
  const int g    = blockIdx.x;
  const bool gemm = (g < NGEMM);
  const float noise_scale = 1.4142135623730951e-3f;  // sqrt(2*tau*sigma_rec^2)

  // ---------------- init phase ----------------
  if (gemm) {
    const int j0 = g * 16;
    // Stage A = W_eff^T tile: A[m=j_local][k=i] = |W_rec[i][j]| * mask[i]
    // Per WMMA chunk c, lane L: M = L&15, K = 4c + 2*(L>>4) + {0,1}  (ds_load_b64)
    for (int idx = tid; idx < 128 * 32; idx += 256) {
      const int c = idx >> 5, L = idx & 31;
      const int j = j0 + (L & 15);
      const int k0 = 4 * c + 2 * (L >> 4);
      v2f v; v[0] = 0.f; v[1] = 0.f;
      if (j < HDIM) {
        if (k0     < HDIM) v[0] = fabsf(W_rec[k0 * HDIM + j])       * ((k0     < 400) ? 1.f : -1.f);
        if (k0 + 1 < HDIM) v[1] = fabsf(W_rec[(k0 + 1) * HDIM + j]) * ((k0 + 1 < 400) ? 1.f : -1.f);
      }
      lds_A[idx] = v;
    }
    if (tid < 16) {
      const int j = j0 + tid;
      lds_win0[tid] = (j < HDIM) ? W_in[j] : 0.f;
      lds_win1[tid] = (j < HDIM) ? W_in[HDIM + j] : 0.f;
      lds_brec[tid] = (j < HDIM) ? b_rec[j] : 0.f;
    }
    // hT buffers: h_0 transposed into buf0; buf1 zeroed (pad rows stay 0 forever)
    for (int idx = tid; idx < 16 * BATCH; idx += 256) {
      const int jl = idx >> 7, b = idx & 127;
      const int j = j0 + jl;
      hT0[j * BATCH + b] = (j < HDIM) ? hidden[b * HDIM + j] : 0.f;
      hT1[j * BATCH + b] = 0.f;
    }
  }
  grid_barrier(bar, 1u);

  const int lane = tid & 31;
  const int wave = tid >> 5;
  const int hi   = lane >> 4;
  const int li   = lane & 15;

  if (gemm) {
    // ---------------- recurrent GEMM workgroup ----------------
    const int j0 = g * 16;
    const int b  = wave * 16 + li;  // batch column owned by this lane (N dim)
    const unsigned lds_off = (unsigned)(uintptr_t)(void*)&lds_h[0];

    for (int t = 0; t < TSTEPS; ++t) {
      const float* cur = (t & 1) ? hT1 : hT0;   // h_t
      float* nxt       = (t & 1) ? hT0 : hT1;   // h_{t+1}
      if (tid == 0) {
        tdm_load_hT(lds_off, cur);
        __builtin_amdgcn_s_wait_tensorcnt(0);
      }
      __syncthreads();

      // C^T tile (16 hidden-out x 16 batch) = W_eff^T x hT, K = 512 in chunks of 4
      v8f acc = {0.f, 0.f, 0.f, 0.f, 0.f, 0.f, 0.f, 0.f};
      #pragma unroll 4
      for (int c = 0; c < 128; ++c) {
        const v2f a = lds_A[c * 32 + lane];
        const int kb = 4 * c + 2 * hi;          // B: VGPR0=K{kb}, VGPR1=K{kb+1}
        v2f bv;
        bv[0] = lds_h[kb * LDSROW + b];
        bv[1] = lds_h[(kb + 1) * LDSROW + b];
        acc = __builtin_amdgcn_wmma_f32_16x16x4_f32(false, a, false, bv,
                                                    (short)0, acc, false, false);
      }

      // elementwise: h_{t+1} = relu((-h + xW_in + hW_eff + b_rec + ns*n)/tau)
      const v2f in2 = *(const v2f*)(input + ((size_t)b * TSTEPS + t) * 2);
      const int jbase = j0 + 8 * hi;
      const size_t nbase = ((size_t)b * TSTEPS + t) * HDIM;
      float nz[8];
      if (j0 + 16 <= HDIM) {         // wave-uniform (blockIdx-only): fast vector path
        const v4f* np = (const v4f*)(noise + nbase + jbase);
        const v4f n0 = np[0], n1 = np[1];
        nz[0]=n0[0]; nz[1]=n0[1]; nz[2]=n0[2]; nz[3]=n0[3];
        nz[4]=n1[0]; nz[5]=n1[1]; nz[6]=n1[2]; nz[7]=n1[3];
      } else {                       // last col-tile: branchless clamped loads
        #pragma unroll
        for (int r = 0; r < 8; ++r) {
          int jg = jbase + r;
          jg = (jg < HDIM) ? jg : (HDIM - 1);   // clamped value unused when jg>=HDIM
          nz[r] = noise[nbase + jg];
        }
      }
      #pragma unroll
      for (int r = 0; r < 8; ++r) {
        const int jl16 = r + 8 * hi;            // C layout: M = r + 8*(lane>=16)
        const int jg = j0 + jl16;
        const float hold = lds_h[jg * LDSROW + b];
        const float xw = in2[0] * lds_win0[jl16] + in2[1] * lds_win1[jl16];
        const float pre = (-hold + xw + acc[r] + lds_brec[jl16]
                           + noise_scale * nz[r]) / 0.01f;
        const float hnew = pre > 0.f ? pre : 0.f;
        if (jg < HDIM) nxt[jg * BATCH + b] = hnew;
      }
      grid_barrier(bar, (unsigned)(t + 2));
    }
    // h_final: h_500 lives in buf0 (500 even); de-transpose into d_out tail
    for (int idx = tid; idx < 16 * BATCH; idx += 256) {
      const int jl = idx & 15, bb = idx >> 4;
      const int j = j0 + jl;
      if (j < HDIM) out[BATCH * TSTEPS + bb * HDIM + j] = hT0[j * BATCH + bb];
    }
  } else {
    // ---------------- output workgroup: out[b,t] = h_{t+1} . W_out + b_out ----
    const int b = tid & 127, s = tid >> 7;
    const float bo = b_out[0];
    for (int t = 0; t < TSTEPS; ++t) {
      if (t >= 1) {
        const float* hT = (t & 1) ? hT1 : hT0;  // h_t lives in buf[t&1]
        const int jb = s ? 256 : 0;
        const int je = s ? HDIM : 256;
        float sum = 0.f;
        #pragma unroll 4
        for (int j = jb; j < je; ++j) sum += hT[j * BATCH + b] * W_out[j];
        lds_red[tid] = sum;
        __syncthreads();
        if (s == 0) out[b * TSTEPS + (t - 1)] = lds_red[tid] + lds_red[tid + 128] + bo;
        __syncthreads();
      }
      grid_barrier(bar, (unsigned)(t + 2));
    }
    {   // out_{T-1} from h_500 (buf0)
      const int jb = s ? 256 : 0;
      const int je = s ? HDIM : 256;
      float sum = 0.f;
      #pragma unroll 4
      for (int j = jb; j < je; ++j) sum += hT0[j * BATCH + b] * W_out[j];
      lds_red[tid] = sum;
      __syncthreads();
      if (s == 0) out[b * TSTEPS + (TSTEPS - 1)] = lds_red[tid] + lds_red[tid + 128] + bo;
    }
  }
}

extern "C" void kernel_launch(void* const* d_in, const int* in_sizes, int n_in,
                              void* d_out, int out_size, void* d_ws, size_t ws_size,
                              hipStream_t stream) {
  const float* input  = (const float*)d_in[0];
  const float* hidden = (const float*)d_in[1];
  const float* noise  = (const float*)d_in[2];
  const float* W_rec  = (const float*)d_in[3];
  const float* W_in   = (const float*)d_in[4];
  const float* b_rec  = (const float*)d_in[5];
  const float* W_out  = (const float*)d_in[6];
  const float* b_out  = (const float*)d_in[7];
  float* out = (float*)d_out;

  unsigned* bar = (unsigned*)d_ws;                       // barrier counter
  float* hT0 = (float*)((char*)d_ws + 256);              // 512*128 f32
  float* hT1 = hT0 + HP * BATCH;                         // 512*128 f32

  (void)hipMemsetAsync(d_ws, 0, 256, stream);            // zero barrier each call

  zrnn_kernel<<<dim3(NBLK), dim3(256), 0, stream>>>(
      input, hidden, noise, W_rec, W_in, b_rec, W_out, b_out,
      out, bar, hT0, hT1);
}